// NMS_15728170238047
// MI455X (gfx1250) — compile-verified
//
#include <hip/hip_runtime.h>

// ---------------- problem constants ----------------
#define NBATCH   8
#define NPRED    25200
#define NPAD     32768        // next pow2 >= NPRED, sorted fully in 256KB LDS
#define NCLS     80
#define NFEAT    85
#define KTOP     1000
#define KPAD     1024
#define CONF_T   0.25f
#define IOU_T    0.45f
#define MAX_WH   4096.0f

typedef __attribute__((ext_vector_type(2))) float v2f;
typedef __attribute__((ext_vector_type(8))) float v8f;
typedef unsigned long long u64;
typedef unsigned int u32;

// float -> order-preserving uint (ascending float == ascending uint)
__device__ __forceinline__ u32 f2ord(float f) {
  u32 u = __float_as_uint(f);
  return (u & 0x80000000u) ? ~u : (u | 0x80000000u);
}
__device__ __forceinline__ float ord2f(u32 u) {
  return __uint_as_float((u & 0x80000000u) ? (u & 0x7FFFFFFFu) : ~u);
}

// ---------------------------------------------------------------------------
// Kernel 1: decode.  score = obj * max(cls); box xywh->xyxy; 64-bit sort key.
// key = (ord(masked_score) << 32) | (0xFFFFFFFF - index)  -> descending sort
// reproduces jax.lax.top_k order incl. lower-index-first tie break.
// ---------------------------------------------------------------------------
__global__ __launch_bounds__(256) void nms_decode(
    const float* __restrict__ x, float* __restrict__ boxes,
    u32* __restrict__ clsid, u64* __restrict__ keys) {
  int t = blockIdx.x * 256 + threadIdx.x;
  int b = t >> 15;          // / NPAD
  int i = t & (NPAD - 1);
  if (b >= NBATCH) return;
  size_t kidx = (size_t)b * NPAD + i;
  if (i >= NPRED) { keys[kidx] = 0ull; return; }   // pad -> smallest key

  const float* p = x + ((size_t)b * NPRED + i) * NFEAT;
  __builtin_prefetch(p + 40, 0, 0);    // gfx1250 global_prefetch_b8
  float cx = p[0], cy = p[1], w = p[2], h = p[3], obj = p[4];
  float cmax = p[5];
  int cid = 0;
#pragma unroll 4
  for (int c = 1; c < NCLS; ++c) {
    float v = p[5 + c];
    if (v > cmax) { cmax = v; cid = c; }   // first-max, matches jnp.argmax
  }
  float score  = obj * cmax;
  float masked = (score > CONF_T) ? score : -1.0f;
  keys[kidx] = ((u64)f2ord(masked) << 32) | (u64)(0xFFFFFFFFu - (u32)i);

  size_t bi = (size_t)b * NPRED + i;
  // w*0.5f is exact (power-of-two scale), so sub/add round identically to ref
  float4 bx;
  bx.x = cx - w * 0.5f; bx.y = cy - h * 0.5f;
  bx.z = cx + w * 0.5f; bx.w = cy + h * 0.5f;
  ((float4*)boxes)[bi] = bx;
  clsid[bi] = (u32)cid;
}

// ---------------------------------------------------------------------------
// Kernel 2: exact descending bitonic sort of 32768 keys per batch, entirely
// in 256KB LDS (CDNA5: 320KB per WGP).  One workgroup (1024 threads) / batch.
// Pair-enumerated exchange loop: 16 pairs/thread/pass, no idle lanes.
// Sorted top-KPAD keys written back in place at keys[b*NPAD + 0..KPAD-1].
// ---------------------------------------------------------------------------
__global__ __launch_bounds__(1024) void nms_sort(u64* __restrict__ keys) {
  extern __shared__ u64 k[];            // NPAD * 8 bytes = 256KB
  const int tid = threadIdx.x;
  const size_t base = (size_t)blockIdx.x * NPAD;

  for (int e = tid; e < NPAD; e += 1024) k[e] = keys[base + e];
  __syncthreads();

  for (int kk = 2; kk <= NPAD; kk <<= 1) {
    for (int jj = kk >> 1; jj > 0; jj >>= 1) {
      for (int p = tid; p < (NPAD >> 1); p += 1024) {
        int i   = ((p & ~(jj - 1)) << 1) | (p & (jj - 1));
        int ixj = i | jj;
        u64 a = k[i], c = k[ixj];
        bool desc = ((i & kk) == 0);
        if (desc ? (a < c) : (a > c)) { k[i] = c; k[ixj] = a; }
      }
      __syncthreads();
    }
  }
  if (tid < KPAD) keys[base + tid] = k[tid];
}

// ---------------------------------------------------------------------------
// Kernel 3: fused NMS per batch (one 1024-thread workgroup / WGP).
//  A) gather top-K boxes/score/class into LDS, class-offset boxes, areas.
//  B) build 1024x1024 (iou > IOU_T) bitmask in LDS.  Pairwise denominator
//     area[i]+area[j] computed as a rank-2 matmul on the matrix unit:
//     v_wmma_f32_16x16x4_f32 with A=[area_i | 1], B=[1 ; area_j].
//     Union/divide kept contraction-free to match the reference rounding.
//  C) serial greedy suppression scan over LDS bitmask rows (ref semantics).
//  D) emit det rows (zeros where not kept).
// ---------------------------------------------------------------------------
__global__ __launch_bounds__(1024) void nms_scan(
    const u64* __restrict__ keys, const float* __restrict__ boxes,
    const u32* __restrict__ clsid, float* __restrict__ out) {
  extern __shared__ char smem[];
  u32*   mask  = (u32*)smem;                       // 1024*32 words = 128KB
  float* ob0   = (float*)(smem + 131072);          // offset boxes (SoA)
  float* ob1   = ob0 + KPAD;
  float* ob2   = ob1 + KPAD;
  float* ob3   = ob2 + KPAD;
  float* bx0   = ob3 + KPAD;                       // original boxes (SoA)
  float* bx1   = bx0 + KPAD;
  float* bx2   = bx1 + KPAD;
  float* bx3   = bx2 + KPAD;
  float* areaS = bx3 + KPAD;
  float* scorS = areaS + KPAD;
  float* clsS  = scorS + KPAD;
  u32*   valS  = (u32*)(clsS + KPAD);
  u32*   keepS = valS + KPAD;
  u32*   sup   = keepS + KPAD;                     // 32 words
  int*   kflag = (int*)(sup + 32);

  const int tid = threadIdx.x;
  const int b   = blockIdx.x;

  // ---- Phase A: gather ----
  u64 key = keys[(size_t)b * NPAD + tid];
  u32 su  = (u32)(key >> 32);
  float scoreV = ord2f(su);                        // top_scores[tid]
  u32 idx = 0xFFFFFFFFu - (u32)(key & 0xFFFFFFFFu);
  bool slot = (tid < KTOP);

  float4 bx = {0.f, 0.f, 0.f, 0.f};
  u32 cid = 0;
  if (slot && idx < (u32)NPRED) {
    bx  = ((const float4*)boxes)[(size_t)b * NPRED + idx];
    cid = clsid[(size_t)b * NPRED + idx];
  }
  float cf  = (float)cid;
  float off = cf * MAX_WH;                         // exact: int * 2^12
  float o0 = bx.x + off, o1 = bx.y + off, o2 = bx.z + off, o3 = bx.w + off;
  ob0[tid] = o0; ob1[tid] = o1; ob2[tid] = o2; ob3[tid] = o3;
  bx0[tid] = bx.x; bx1[tid] = bx.y; bx2[tid] = bx.z; bx3[tid] = bx.w;
  areaS[tid] = (o2 - o0) * (o3 - o1);
  scorS[tid] = scoreV;
  clsS[tid]  = cf;
  valS[tid]  = (slot && (scoreV > CONF_T)) ? 1u : 0u;
  keepS[tid] = 0u;
  if (tid < 32) sup[tid] = 0u;
  for (int w = tid; w < KPAD * 32; w += 1024) mask[w] = 0u;
  __syncthreads();

  // ---- Phase B: bitmask via WMMA denominator tiles ----
  const int wid  = tid >> 5;
  const int lane = tid & 31;
  const int l16  = lane & 15;
  const bool lo  = lane < 16;

  for (int tile = wid; tile < 64 * 64; tile += 32) {
    int TI = tile >> 6;        // row tile (i dim)
    int TJ = tile & 63;        // col tile (j dim)

    // A (16x4): lanes 0-15 hold K0,K1 for M=lane; lanes 16-31 hold K2,K3 (=0)
    v2f a;  a.x = lo ? areaS[TI * 16 + l16] : 0.0f;
            a.y = lo ? 1.0f : 0.0f;
    // B (4x16): VGPR0 = K0 row (lanes 0-15) / K2 (lanes 16-31),
    //           VGPR1 = K1 row / K3.   K0 = ones, K1 = area_j.
    v2f bm; bm.x = lo ? 1.0f : 0.0f;
            bm.y = lo ? areaS[TJ * 16 + l16] : 0.0f;
    v8f c = {};
    // D[m][n] = area_i[m] + area_j[n]   (matrix unit; products exact so the
    // single-rounded sum equals the reference's v_add)
    v8f d = __builtin_amdgcn_wmma_f32_16x16x4_f32(
        false, a, false, bm, (short)0, c, false, false);

    int nIdx  = TJ * 16 + l16;
    int mBase = TI * 16 + (lo ? 0 : 8);
    float jx1 = ob0[nIdx], jy1 = ob1[nIdx], jx2 = ob2[nIdx], jy2 = ob3[nIdx];
#pragma unroll
    for (int v = 0; v < 8; ++v) {
#pragma clang fp contract(off)
      int m = mBase + v;
      float x1 = fmaxf(ob0[m], jx1);
      float y1 = fmaxf(ob1[m], jy1);
      float x2 = fminf(ob2[m], jx2);
      float y2 = fminf(ob3[m], jy2);
      float iw = fmaxf(x2 - x1, 0.0f);
      float ih = fmaxf(y2 - y1, 0.0f);
      float inter = iw * ih;                 // rounded mul (no fma fusion)
      float uni   = d[v] - inter;            // rounded sub, matches reference
      uni = uni + 1e-9f;                     // rounded add
      float iou = inter / uni;               // IEEE divide == XLA div
      if ((iou > IOU_T) && (m != nIdx))
        atomicOr(&mask[m * 32 + (nIdx >> 5)], 1u << (nIdx & 31));
    }
  }
  __syncthreads();

  // ---- Phase C: serial greedy scan (matches jax.lax.scan step) ----
  for (int i = 0; i < KTOP; ++i) {
    if (tid == 0) {
      bool s = (sup[i >> 5] >> (i & 31)) & 1u;
      int kp = (!s && valS[i]) ? 1 : 0;
      keepS[i] = (u32)kp;
      *kflag = kp;
    }
    __syncthreads();
    if (*kflag && tid < 32) sup[tid] |= mask[i * 32 + tid];
    __syncthreads();
  }

  // ---- Phase D: output det rows ----
  if (tid < KTOP) {
    bool kp = keepS[tid] != 0u;
    size_t o = ((size_t)b * KTOP + tid) * 6;
    out[o + 0] = kp ? bx0[tid] : 0.0f;
    out[o + 1] = kp ? bx1[tid] : 0.0f;
    out[o + 2] = kp ? bx2[tid] : 0.0f;
    out[o + 3] = kp ? bx3[tid] : 0.0f;
    out[o + 4] = kp ? scorS[tid] : 0.0f;
    out[o + 5] = kp ? clsS[tid] : 0.0f;
  }
}

// ---------------------------------------------------------------------------
extern "C" void kernel_launch(void* const* d_in, const int* in_sizes, int n_in,
                              void* d_out, int out_size, void* d_ws, size_t ws_size,
                              hipStream_t stream) {
  const float* x = (const float*)d_in[0];
  float* out = (float*)d_out;

  // workspace layout (~6.1 MB total)
  char* w = (char*)d_ws;
  const size_t BOX_BYTES = (size_t)NBATCH * NPRED * 4 * sizeof(float); // 3,225,600
  const size_t CLS_BYTES = (size_t)NBATCH * NPRED * sizeof(u32);       //   806,400
  float* boxes = (float*)w;
  u32*   clsid = (u32*)(w + BOX_BYTES);
  u64*   keys  = (u64*)(w + BOX_BYTES + CLS_BYTES);                    // 2,097,152

  nms_decode<<<(NBATCH * NPAD) / 256, 256, 0, stream>>>(x, boxes, clsid, keys);

  nms_sort<<<NBATCH, 1024, NPAD * sizeof(u64), stream>>>(keys);        // 256KB LDS

  const size_t LDS3 = 131072                       // mask bits
                    + (size_t)KPAD * 13 * 4        // ob*4, bx*4, area, score, cls, val, keep
                    + 32 * 4 + 16;                 // sup + kflag (+pad)
  nms_scan<<<NBATCH, 1024, LDS3, stream>>>(keys, boxes, clsid, out);
}